// SPE_89429809037686
// MI455X (gfx1250) — compile-verified
//
#include <hip/hip_runtime.h>
#include <hip/hip_bf16.h>

typedef __attribute__((ext_vector_type(2))) float v2f;
typedef __attribute__((ext_vector_type(4))) float v4f;
typedef __attribute__((ext_vector_type(8))) float v8f;

#define BATCH 2
#define CCH   64        // C_in == C_out == 64
#define NPTS  40960
#define KNBR  16
#define BN_EPS 1e-5f

// ---------------------------------------------------------------------------
// Kernel 1: transpose feature (B, C, N) -> feat_t (B, N, C) so each gathered
// point is one contiguous 256B row. Classic 32x32 LDS tile transpose.
// ---------------------------------------------------------------------------
__global__ __launch_bounds__(256) void SPE_transpose_kernel(
    const float* __restrict__ feat, float* __restrict__ feat_t) {
  __shared__ float tile[32][33];
  const int bn = blockIdx.x * 32;          // n tile
  const int bc = blockIdx.y * 32;          // c tile
  const int b  = blockIdx.z;
  const int tx = threadIdx.x;              // 0..31
  const int ty = threadIdx.y;              // 0..7

  const float* src = feat + ((size_t)b * CCH + bc) * NPTS + bn;
  #pragma unroll
  for (int i = ty; i < 32; i += 8)
    tile[i][tx] = src[(size_t)i * NPTS + tx];      // coalesced in n
  __syncthreads();
  float* dst = feat_t + ((size_t)b * NPTS + bn) * CCH + bc;
  #pragma unroll
  for (int i = ty; i < 32; i += 8)
    dst[(size_t)i * CCH + tx] = tile[tx][i];       // coalesced in c
}

// ---------------------------------------------------------------------------
// Kernel 2: gather + WMMA GEMM + BN + ReLU + sum over K neighbors.
//
// K-slice -> channel mapping chosen as c(ch,k) = 2*ch + (k&1) + 32*(k>>1).
// With the ISA A layout (lane = m + 16*(k>>1), vgpr j -> k = 2*half + j) this
// makes each lane's A data a CONTIGUOUS 32-float span of the gathered row:
//   feat_t[idx[m]][32*half .. 32*half+31]        -> 8x global_load_b128.
// The mirrored B layout gives each lane a contiguous 32-float span of one w
// row: w[t*16+ocol][32*half .. +31]              -> hoisted to registers once.
// Inner loop: 16 K=4 slices x 4 independent accumulators (4-way WMMA ILP),
// no LDS in the MAC loop. Epilogue: BN affine + ReLU, in-register row sum,
// shfl_xor(16) cross-half add, stage [64 o][64 n] tile in LDS, coalesced out.
// ---------------------------------------------------------------------------
__global__ __launch_bounds__(256) void SPE_gconv_wmma_kernel(
    const float* __restrict__ feat_t, const int* __restrict__ nidx,
    const float* __restrict__ w, const float* __restrict__ gamma,
    const float* __restrict__ beta, const float* __restrict__ rmean,
    const float* __restrict__ rvar, float* __restrict__ out) {
  __shared__ __align__(16) float lds_out[CCH * 64];    // 16 KB: [o][n_local]

  const int b    = blockIdx.y;
  const int n0   = blockIdx.x * 64;
  const int tid  = threadIdx.x;
  const int lane = tid & 31;
  const int wave = tid >> 5;
  const int ocol = lane & 15;      // output column within a 16-wide tile
  const int half = lane >> 4;      // lane half -> K-pair select

  // Hoisted B fragments (loop-invariant): 32 contiguous floats per tile.
  v4f Braw[4][8];
  #pragma unroll
  for (int t = 0; t < 4; ++t) {
    const float* wrow = w + (size_t)(t * 16 + ocol) * CCH + 32 * half;
    #pragma unroll
    for (int q = 0; q < 8; ++q) Braw[t][q] = *(const v4f*)(wrow + 4 * q);
  }

  // Per-lane BN affine params for the 4 output tiles this lane touches.
  float scl[4], sft[4];
  #pragma unroll
  for (int t = 0; t < 4; ++t) {
    const int o = t * 16 + ocol;
    const float s = gamma[o] * rsqrtf(rvar[o] + BN_EPS);
    scl[t] = s;
    sft[t] = beta[o] - rmean[o] * s;
  }

  // 8 waves x 8 n each = 64 n per workgroup.
  for (int ni = 0; ni < 8; ++ni) {
    const int n = n0 + wave * 8 + ni;
    const int m = lane & 15;                             // neighbor row index
    int idxm = nidx[((size_t)b * NPTS + n) * KNBR + m];
    idxm = idxm < 0 ? 0 : (idxm > NPTS - 1 ? NPTS - 1 : idxm);

    // Gather: 8x b128 = this lane's contiguous 32-channel span.
    const float* arow =
        feat_t + ((size_t)b * NPTS + idxm) * CCH + 32 * half;
    v4f Araw[8];
    #pragma unroll
    for (int q = 0; q < 8; ++q) Araw[q] = *(const v4f*)(arow + 4 * q);

    // MAC: 16 slices x 4 independent accumulator chains.
    v8f acc[4];
    #pragma unroll
    for (int t = 0; t < 4; ++t) acc[t] = (v8f){};
    #pragma unroll
    for (int ch = 0; ch < 16; ++ch) {
      const int q = ch >> 1, e = 2 * (ch & 1);
      v2f Af;
      Af[0] = Araw[q][e];
      Af[1] = Araw[q][e + 1];
      #pragma unroll
      for (int t = 0; t < 4; ++t) {
        v2f Bf;
        Bf[0] = Braw[t][q][e];
        Bf[1] = Braw[t][q][e + 1];
        acc[t] = __builtin_amdgcn_wmma_f32_16x16x4_f32(
            false, Af, false, Bf, (short)0, acc[t], false, false);
      }
    }

    // BN + ReLU + reduce over the 16 neighbor rows.
    #pragma unroll
    for (int t = 0; t < 4; ++t) {
      float s = 0.f;
      #pragma unroll
      for (int r = 0; r < 8; ++r) {
        const float v = acc[t][r] * scl[t] + sft[t];
        s += fmaxf(v, 0.f);
      }
      s += __shfl_xor(s, 16);   // rows 0..7 half + rows 8..15 half
      if (half == 0)
        lds_out[(t * 16 + ocol) * 64 + wave * 8 + ni] = s;
    }
  }
  __syncthreads();

  // Coalesced write-out: out[b, o, n0 + nn], 64 consecutive n per row.
  float* obase = out + (size_t)b * CCH * NPTS + n0;
  for (int i = tid; i < CCH * 64; i += 256) {
    const int o  = i >> 6;
    const int nn = i & 63;
    obase[(size_t)o * NPTS + nn] = lds_out[i];
  }
}

extern "C" void kernel_launch(void* const* d_in, const int* in_sizes, int n_in,
                              void* d_out, int out_size, void* d_ws, size_t ws_size,
                              hipStream_t stream) {
  const float* feature = (const float*)d_in[0];   // (B, C, N)
  const int*   nidx    = (const int*)d_in[1];     // (B, N, K)
  const float* w       = (const float*)d_in[2];   // (C_out, C_in)
  const float* gamma   = (const float*)d_in[3];
  const float* beta    = (const float*)d_in[4];
  const float* rmean   = (const float*)d_in[5];
  const float* rvar    = (const float*)d_in[6];
  float* out = (float*)d_out;                     // (B, C_out, N, 1)

  float* feat_t = (float*)d_ws;                   // (B, N, C) staging, 21 MB

  dim3 tgrid(NPTS / 32, CCH / 32, BATCH);
  dim3 tblk(32, 8, 1);
  SPE_transpose_kernel<<<tgrid, tblk, 0, stream>>>(feature, feat_t);

  dim3 ggrid(NPTS / 64, BATCH, 1);
  SPE_gconv_wmma_kernel<<<ggrid, 256, 0, stream>>>(
      feat_t, nidx, w, gamma, beta, rmean, rvar, out);
}